// LightGCN_2_66185446031940
// MI455X (gfx1250) — compile-verified
//
#include <hip/hip_runtime.h>
#include <hip/hip_bf16.h>
#include <cstdint>
#include <cstddef>

// ---------------------------------------------------------------------------
// LightGCN forward on MI455X (gfx1250).
//   h0 = emb[x];  h1 = A @ drop(h0,u1);  h2 = A @ drop(h1,u2);  out=(h0+h1+h2)/3
// Random COO sparsity (1 nz per ~16x16 tile) => WMMA is counterproductive.
// The win on CDNA5: h (51MB) lives in the 192MB L2, so SpMM = L2 gathers +
// hardware f32 atomics (global_atomic_add_f32), with edge lists staged into
// LDS via async global->LDS loads (ASYNCcnt path) and wave-uniform edge
// scalars promoted to SGPRs (readfirstlane) for saddr-form addressing.
// ---------------------------------------------------------------------------

#define RATE        0.1f
#define KEEPSCALE   (1.0f / (1.0f - RATE))
#define INV3        (1.0f / 3.0f)
#define DV4         32          // D/4 = 128/4 float4 per row
#define EDGES_PER_BLOCK 256

// ---- async global -> LDS staging (gfx1250), with portable fallback --------
#if defined(__has_builtin)
#if __has_builtin(__builtin_amdgcn_global_load_async_to_lds_b32)
#define HAVE_ASYNC_LDS 1
#endif
#endif

#ifdef HAVE_ASYNC_LDS
typedef __attribute__((address_space(1))) int* gas_i32_ptr;
typedef __attribute__((address_space(3))) int* las_i32_ptr;

__device__ __forceinline__ void async_copy_b32(const void* gsrc, void* lds_dst) {
  __builtin_amdgcn_global_load_async_to_lds_b32(
      (gas_i32_ptr)(uintptr_t)gsrc,
      (las_i32_ptr)lds_dst,
      /*offset=*/0, /*cpol=*/0);
}
__device__ __forceinline__ void wait_async0() {
#if __has_builtin(__builtin_amdgcn_s_wait_asynccnt)
  __builtin_amdgcn_s_wait_asynccnt(0);
#else
  asm volatile("s_wait_asynccnt 0" ::: "memory");
#endif
}
#endif

// wave-uniform scalar promotion (value is identical in all lanes)
__device__ __forceinline__ int rfl_i32(int v) {
  return __builtin_amdgcn_readfirstlane(v);
}
__device__ __forceinline__ float rfl_f32(float v) {
  return __int_as_float(__builtin_amdgcn_readfirstlane(__float_as_int(v)));
}

// ---- kernels --------------------------------------------------------------

// tmp = dropout(emb[x], u1)        (vectorized float4, n4 = N*D/4)
__global__ void __launch_bounds__(256)
k_embed_dropout(const int* __restrict__ x, const float4* __restrict__ emb4,
                const float4* __restrict__ u4, float4* __restrict__ tmp4, int n4)
{
  int idx = blockIdx.x * 256 + threadIdx.x;
  if (idx >= n4) return;
  int row = idx >> 5;           // / DV4
  int col = idx & (DV4 - 1);
  int src = x[row];
  float4 e = emb4[(size_t)src * DV4 + col];
  float4 u = u4[idx];
  float4 o;
  o.x = (u.x >= RATE) ? e.x * KEEPSCALE : 0.0f;
  o.y = (u.y >= RATE) ? e.y * KEEPSCALE : 0.0f;
  o.z = (u.z >= RATE) ? e.z * KEEPSCALE : 0.0f;
  o.w = (u.w >= RATE) ? e.w * KEEPSCALE : 0.0f;
  tmp4[idx] = o;
}

__global__ void __launch_bounds__(256)
k_zero(float4* __restrict__ p, int n4)
{
  int idx = blockIdx.x * 256 + threadIdx.x;
  if (idx < n4) p[idx] = make_float4(0.f, 0.f, 0.f, 0.f);
}

// COO SpMM: out[rows[e]] += vals[e] * h[cols[e]]     (wave32 per edge)
// Edge triples staged to LDS via async-to-LDS; promoted to SGPRs; per-lane
// float4 gather from L2; 4x global_atomic_add_f32 scatter.
__global__ void __launch_bounds__(256)
k_spmm(const int* __restrict__ rows, const int* __restrict__ cols,
       const float* __restrict__ vals, const float4* __restrict__ h4,
       float* __restrict__ out, int E)
{
  __shared__ int   s_rows[EDGES_PER_BLOCK];
  __shared__ int   s_cols[EDGES_PER_BLOCK];
  __shared__ float s_vals[EDGES_PER_BLOCK];

  const int t    = threadIdx.x;
  const int base = blockIdx.x * EDGES_PER_BLOCK;
  int g = base + t;
  if (g >= E) g = E - 1;        // clamp: stages a valid (unused) edge for tails

#ifdef HAVE_ASYNC_LDS
  async_copy_b32(&rows[g], &s_rows[t]);
  async_copy_b32(&cols[g], &s_cols[t]);
  async_copy_b32(&vals[g], &s_vals[t]);
  wait_async0();
#else
  s_rows[t] = rows[g];
  s_cols[t] = cols[g];
  s_vals[t] = vals[g];
#endif
  __syncthreads();

  const int wave   = t >> 5;
  const int lane   = t & 31;
  const int nEdges = min(EDGES_PER_BLOCK, E - base);

  int i = wave * 32;
  const int iEnd = min(i + 32, nEdges);
#pragma unroll 4
  for (; i < iEnd; ++i) {
    // wave-uniform edge scalars -> SGPRs: saddr-form addressing, SALU row math
    const int   c = rfl_i32(s_cols[i]);
    const int   r = rfl_i32(s_rows[i]);
    const float v = rfl_f32(s_vals[i]);
    const float4 hv = h4[(size_t)c * DV4 + lane];     // global_load_b128 (L2 hit)
    float* dst = out + (size_t)r * 128 + lane * 4;
    unsafeAtomicAdd(dst + 0, v * hv.x);               // global_atomic_add_f32
    unsafeAtomicAdd(dst + 1, v * hv.y);
    unsafeAtomicAdd(dst + 2, v * hv.z);
    unsafeAtomicAdd(dst + 3, v * hv.w);
  }
}

// Fused: out = (emb[x] + h1) / 3   AND   tmp = dropout(h1, u2)   (reads h1 once)
__global__ void __launch_bounds__(256)
k_fuse(const int* __restrict__ x, const float4* __restrict__ emb4,
       const float4* __restrict__ h1_4, const float4* __restrict__ u4,
       float4* __restrict__ out4, float4* __restrict__ tmp4, int n4)
{
  int idx = blockIdx.x * 256 + threadIdx.x;
  if (idx >= n4) return;
  int row = idx >> 5;
  int col = idx & (DV4 - 1);
  int src = x[row];
  float4 h = h1_4[idx];
  float4 e = emb4[(size_t)src * DV4 + col];
  float4 u = u4[idx];

  out4[idx] = make_float4((e.x + h.x) * INV3, (e.y + h.y) * INV3,
                          (e.z + h.z) * INV3, (e.w + h.w) * INV3);
  float4 d;
  d.x = (u.x >= RATE) ? h.x * KEEPSCALE : 0.0f;
  d.y = (u.y >= RATE) ? h.y * KEEPSCALE : 0.0f;
  d.z = (u.z >= RATE) ? h.z * KEEPSCALE : 0.0f;
  d.w = (u.w >= RATE) ? h.w * KEEPSCALE : 0.0f;
  tmp4[idx] = d;
}

// out += h2 / 3
__global__ void __launch_bounds__(256)
k_final(const float4* __restrict__ h2_4, float4* __restrict__ out4, int n4)
{
  int idx = blockIdx.x * 256 + threadIdx.x;
  if (idx >= n4) return;
  float4 o = out4[idx];
  float4 h = h2_4[idx];
  o.x += h.x * INV3;  o.y += h.y * INV3;
  o.z += h.z * INV3;  o.w += h.w * INV3;
  out4[idx] = o;
}

// ---- launch ---------------------------------------------------------------
extern "C" void kernel_launch(void* const* d_in, const int* in_sizes, int n_in,
                              void* d_out, int out_size, void* d_ws, size_t ws_size,
                              hipStream_t stream) {
  const int*   x    = (const int*)d_in[0];
  const int*   rows = (const int*)d_in[1];
  const int*   cols = (const int*)d_in[2];
  const float* vals = (const float*)d_in[3];
  const float* emb  = (const float*)d_in[4];
  const float* u1   = (const float*)d_in[5];
  const float* u2   = (const float*)d_in[6];
  float*       out  = (float*)d_out;

  const int E  = in_sizes[1];
  const int n4 = out_size / 4;                 // N*D/4 float4 elements

  float* tmp = (float*)d_ws;                   // N*D floats: dropout buffer
  float* acc = tmp + (size_t)out_size;         // N*D floats: h1, then reused as h2

  dim3 blk(256);
  dim3 grdElem((unsigned)((n4 + 255) / 256));
  dim3 grdEdge((unsigned)((E + EDGES_PER_BLOCK - 1) / EDGES_PER_BLOCK));

  // layer 1
  k_embed_dropout<<<grdElem, blk, 0, stream>>>(x, (const float4*)emb,
                                               (const float4*)u1, (float4*)tmp, n4);
  k_zero<<<grdElem, blk, 0, stream>>>((float4*)acc, n4);
  k_spmm<<<grdEdge, blk, 0, stream>>>(rows, cols, vals, (const float4*)tmp, acc, E);

  // out = (h0+h1)/3 ; tmp = dropout(h1, u2)
  k_fuse<<<grdElem, blk, 0, stream>>>(x, (const float4*)emb, (const float4*)acc,
                                      (const float4*)u2, (float4*)out, (float4*)tmp, n4);

  // layer 2 (reuse acc as h2)
  k_zero<<<grdElem, blk, 0, stream>>>((float4*)acc, n4);
  k_spmm<<<grdEdge, blk, 0, stream>>>(rows, cols, vals, (const float4*)tmp, acc, E);

  // out += h2/3
  k_final<<<grdElem, blk, 0, stream>>>((const float4*)acc, (float4*)out, n4);
}